// SimpleShader_81844896793060
// MI455X (gfx1250) — compile-verified
//
#include <hip/hip_runtime.h>

// SoftRas-style compositing shader for MI455X (gfx1250).
// Memory-bound (208 B/pixel, ~0.5 FLOP/B): optimize the data path, not FLOPs.
//   * colors (96 B/pixel) staged block-wide into LDS via CDNA5 async copies
//     (global_load_async_to_lds_b128 + s_wait_asynccnt), read back with
//     ds_load_b128 at a padded 112 B pixel stride (4-way-max bank conflicts).
//   * dists/zbuf/pix_to_face stream directly as b128 loads with TH=NT
//     (read-once stream: don't pollute 192 MB L2).
//   * output is a single nontemporal b128 store per pixel.

#define KFRAG 8
#define PIX_STRIDE 112   // 96B colors + 16B pad
#define BLOCK 256

typedef __attribute__((address_space(3))) unsigned char lds_uchar;
typedef __attribute__((ext_vector_type(4))) float fvec4;   // native clang vector:
typedef __attribute__((ext_vector_type(4))) int   ivec4;   // required by NT builtins

__global__ void __launch_bounds__(BLOCK)
softras_kernel(const float* __restrict__ colors,
               const float* __restrict__ dists,
               const float* __restrict__ zbuf,
               const float* __restrict__ background,
               const int*   __restrict__ p2f,
               float*       __restrict__ out,
               int P)
{
    __shared__ __align__(16) unsigned char smem[BLOCK * PIX_STRIDE];

    const int tid        = threadIdx.x;
    const int blockStart = blockIdx.x << 8;
    int rem = P - blockStart;
    const int cnt    = rem < BLOCK ? rem : BLOCK;   // pixels handled by this block
    const int nUnits = cnt * 6;                     // 16B units of colors to stage

    // 32-bit LDS byte offset of smem (addrspace(3) pointers are 32-bit offsets)
    unsigned ldsBase =
        (unsigned)(unsigned long long)(lds_uchar*)&smem[0];
    const unsigned long long gBase =
        (unsigned long long)(colors + (long long)blockStart * (KFRAG * 3));

    // Stage this block's colors tile (cnt * 96 B) into LDS, coalesced:
    // unit idx = u*256 + tid copies global bytes [idx*16, idx*16+16) into
    // LDS at pixel-padded layout p*112 + r*16  (p = idx/6, r = idx%6).
#pragma unroll
    for (int u = 0; u < 6; ++u) {
        int idx = u * BLOCK + tid;
        if (idx < nUnits) {
            int p = idx / 6;
            int r = idx - p * 6;
            unsigned lds = ldsBase + (unsigned)(p * PIX_STRIDE + r * 16);
            unsigned long long g = gBase + (unsigned long long)idx * 16ull;
            asm volatile("global_load_async_to_lds_b128 %0, %1, off"
                         :: "v"(lds), "v"(g) : "memory");
        }
    }
    asm volatile("s_wait_asynccnt 0" ::: "memory");
    __syncthreads();

    if (tid < cnt) {
        const int pixel = blockStart + tid;
        const size_t fb = (size_t)pixel * KFRAG;

        // Per-fragment scalars: 2x b128 each, coalesced (32 B/lane stride),
        // nontemporal (read-once stream).
        const fvec4 d0 = __builtin_nontemporal_load((const fvec4*)(dists + fb));
        const fvec4 d1 = __builtin_nontemporal_load((const fvec4*)(dists + fb + 4));
        const fvec4 z0 = __builtin_nontemporal_load((const fvec4*)(zbuf  + fb));
        const fvec4 z1 = __builtin_nontemporal_load((const fvec4*)(zbuf  + fb + 4));
        const ivec4 f0 = __builtin_nontemporal_load((const ivec4*)(p2f   + fb));
        const ivec4 f1 = __builtin_nontemporal_load((const ivec4*)(p2f   + fb + 4));

        float dk[KFRAG] = {d0.x, d0.y, d0.z, d0.w, d1.x, d1.y, d1.z, d1.w};
        float zk[KFRAG] = {z0.x, z0.y, z0.z, z0.w, z1.x, z1.y, z1.z, z1.w};
        int   fk[KFRAG] = {f0.x, f0.y, f0.z, f0.w, f1.x, f1.y, f1.z, f1.w};

        // Pull this pixel's 24 colors from LDS (6x ds_load_b128).
        const fvec4* crow = (const fvec4*)(smem + tid * PIX_STRIDE);
        fvec4 c4[6];
#pragma unroll
        for (int i = 0; i < 6; ++i) c4[i] = crow[i];
        float col[KFRAG * 3];
#pragma unroll
        for (int i = 0; i < 6; ++i) {
            col[i * 4 + 0] = c4[i].x;
            col[i * 4 + 1] = c4[i].y;
            col[i * 4 + 2] = c4[i].z;
            col[i * 4 + 3] = c4[i].w;
        }

        // sigma = gamma = 1e-4  ->  multiply by 1e4
        const float INV_SIGMA = 1.0e4f;
        const float INV_GAMMA = 1.0e4f;
        const float EPSV      = 1.0e-10f;

        float prob[KFRAG], zinv[KFRAG];
        float alpha = 1.0f;
        float zmax  = EPSV;   // clip(max(z_inv), EPS) with lower bound only
#pragma unroll
        for (int k = 0; k < KFRAG; ++k) {
            bool  m  = fk[k] >= 0;
            // sigmoid(-d/sigma) = 1 / (1 + exp(d/sigma))
            float pr = m ? (1.0f / (1.0f + __expf(dk[k] * INV_SIGMA))) : 0.0f;
            prob[k]  = pr;
            alpha   *= (1.0f - pr);
            float zi = m ? (100.0f - zk[k]) * (1.0f / 99.0f) : 0.0f;
            zinv[k]  = zi;
            zmax     = fmaxf(zmax, zi);
        }

        float wsum = 0.0f, r0 = 0.0f, r1 = 0.0f, r2 = 0.0f;
#pragma unroll
        for (int k = 0; k < KFRAG; ++k) {
            float w = prob[k] * __expf((zinv[k] - zmax) * INV_GAMMA);
            wsum += w;
            r0 += w * col[k * 3 + 0];
            r1 += w * col[k * 3 + 1];
            r2 += w * col[k * 3 + 2];
        }

        float delta = fmaxf(__expf((EPSV - zmax) * INV_GAMMA), EPSV);
        float denom = wsum + delta;
        float inv   = 1.0f / denom;

        float b0 = background[0];
        float b1 = background[1];
        float b2 = background[2];

        fvec4 o;
        o.x = (r0 + delta * b0) * inv;
        o.y = (r1 + delta * b1) * inv;
        o.z = (r2 + delta * b2) * inv;
        o.w = 1.0f - alpha;
        __builtin_nontemporal_store(o, (fvec4*)(out + (size_t)pixel * 4));
    }
}

extern "C" void kernel_launch(void* const* d_in, const int* in_sizes, int n_in,
                              void* d_out, int out_size, void* d_ws, size_t ws_size,
                              hipStream_t stream) {
    const float* colors     = (const float*)d_in[0];
    const float* dists      = (const float*)d_in[1];
    const float* zbuf       = (const float*)d_in[2];
    const float* background = (const float*)d_in[3];
    const int*   p2f        = (const int*)d_in[4];
    float*       out        = (float*)d_out;

    const long long P = (long long)in_sizes[1] / KFRAG;   // N*H*W pixels
    const int blocks = (int)((P + BLOCK - 1) / BLOCK);

    softras_kernel<<<blocks, BLOCK, 0, stream>>>(colors, dists, zbuf,
                                                 background, p2f, out, (int)P);
}